// MoEGate_53910429499972
// MI455X (gfx1250) — compile-verified
//
#include <hip/hip_runtime.h>
#include <hip/hip_bf16.h>

// ---------------------------------------------------------------------------
// MoE gating for MI455X (gfx1250): fp32 router GEMM via V_WMMA_F32_16X16X4_F32
// + fused softmax/top-2.  Bandwidth-bound: 335 MB of activations @ 23.3 TB/s
// => ~14.4 us floor; WMMA supplies the ~47 TFLOP/s fp32 needed to ride it.
// ---------------------------------------------------------------------------

#define N_TOKENS 32768
#define EMBED    2560
#define N_EXP    4
#define TOPK     2

#define WAVES_PER_BLOCK 8
#define BLOCK_THREADS   (WAVES_PER_BLOCK * 32)
#define TOKENS_PER_BLOCK (WAVES_PER_BLOCK * 16)
// padded LDS stride for the weight rows: 2564 % 64 == 4 -> expert rows hit
// different bank groups (lanes 0..3 would otherwise 4-way conflict).
#define WSTRIDE  (EMBED + 4)

typedef __attribute__((ext_vector_type(2))) float v2f;
typedef __attribute__((ext_vector_type(8))) float v8f;

__global__ __launch_bounds__(BLOCK_THREADS)
void moe_gate_wmma_kernel(const float* __restrict__ hidden,
                          const float* __restrict__ weight,
                          float* __restrict__ out) {
    __shared__ __align__(16) float s_w[N_EXP * WSTRIDE];          // ~41 KB router weight
    __shared__ float s_log[WAVES_PER_BLOCK][16][N_EXP];           // per-wave logit spill

    const int tid  = threadIdx.x;
    const int wave = tid >> 5;
    const int lane = tid & 31;
    const int row  = lane & 15;       // M (token in tile) for A; N (column) for C
    const int half = lane >> 4;       // half-wave select
    const int koff = half * 2;        // K pair owned by this half-wave
    const int e    = row & 3;         // expert column this lane feeds into B

    // ---- stage router weight (row-major, padded) into LDS -----------------
    for (int i = tid; i < (N_EXP * EMBED) / 4; i += BLOCK_THREADS) {
        const int ew = i / (EMBED / 4);
        const int k4 = i % (EMBED / 4);
        const float4 v = ((const float4*)weight)[i];
        *(float4*)&s_w[ew * WSTRIDE + k4 * 4] = v;
    }
    __syncthreads();

    // ---- streaming WMMA GEMM over D=2560 ----------------------------------
    const int tile_base = blockIdx.x * TOKENS_PER_BLOCK + wave * 16;
    const float* ap = hidden + (size_t)(tile_base + row) * EMBED + koff;
    const float* wp = &s_w[e * WSTRIDE + koff];

    v8f acc0 = {}; v8f acc1 = {}; v8f acc2 = {}; v8f acc3 = {};

    for (int k = 0; k < EMBED; k += 16) {
        // A: 16x4 f32 tile, one b64 per lane per chunk (streams the row)
        v2f a0 = *(const v2f*)(ap + k);
        v2f a1 = *(const v2f*)(ap + k + 4);
        v2f a2 = *(const v2f*)(ap + k + 8);
        v2f a3 = *(const v2f*)(ap + k + 12);
        // B: 4x16 f32 tile from LDS (same half-wave K split as A)
        v2f b0 = *(const v2f*)(wp + k);
        v2f b1 = *(const v2f*)(wp + k + 4);
        v2f b2 = *(const v2f*)(wp + k + 8);
        v2f b3 = *(const v2f*)(wp + k + 12);

        acc0 = __builtin_amdgcn_wmma_f32_16x16x4_f32(false, a0, false, b0,
                                                     (short)0, acc0, false, false);
        acc1 = __builtin_amdgcn_wmma_f32_16x16x4_f32(false, a1, false, b1,
                                                     (short)0, acc1, false, false);
        acc2 = __builtin_amdgcn_wmma_f32_16x16x4_f32(false, a2, false, b2,
                                                     (short)0, acc2, false, false);
        acc3 = __builtin_amdgcn_wmma_f32_16x16x4_f32(false, a3, false, b3,
                                                     (short)0, acc3, false, false);
    }
    v8f acc = (acc0 + acc1) + (acc2 + acc3);

    // ---- spill the 4 real expert columns of C to LDS ----------------------
    // C layout: VGPR i, lanes 0-15 -> (M=i, N=lane); lanes 16-31 -> (M=8+i).
    if (row < N_EXP) {
        #pragma unroll
        for (int i = 0; i < 8; ++i)
            s_log[wave][i + 8 * half][row] = acc[i];
    }
    __syncthreads();

    // ---- softmax + top-2, one token per lane (lanes 0..15) ----------------
    if (lane < 16) {
        const int token = tile_base + lane;
        float l[N_EXP];
        #pragma unroll
        for (int j = 0; j < N_EXP; ++j) l[j] = s_log[wave][lane][j];

        float m = l[0];
        #pragma unroll
        for (int j = 1; j < N_EXP; ++j) m = fmaxf(m, l[j]);

        float p[N_EXP];
        float s = 0.0f;
        #pragma unroll
        for (int j = 0; j < N_EXP; ++j) { p[j] = __expf(l[j] - m); s += p[j]; }
        const float inv = 1.0f / s;
        #pragma unroll
        for (int j = 0; j < N_EXP; ++j) p[j] *= inv;

        // top-2 with lowest-index-wins on ties (matches jax.lax.top_k)
        int i0 = 0;
        #pragma unroll
        for (int j = 1; j < N_EXP; ++j) if (p[j] > p[i0]) i0 = j;
        int i1 = (i0 == 0) ? 1 : 0;
        #pragma unroll
        for (int j = 0; j < N_EXP; ++j)
            if (j != i0 && p[j] > p[i1]) i1 = j;

        int*   out_idx = reinterpret_cast<int*>(out);
        float* out_w   = out + 2 * N_TOKENS;
        int*   out_row = reinterpret_cast<int*>(out) + 4 * N_TOKENS;

        out_idx[token * TOPK + 0] = i0;
        out_idx[token * TOPK + 1] = i1;
        out_w[token * TOPK + 0]   = p[i0];
        out_w[token * TOPK + 1]   = p[i1];
        out_row[token * TOPK + 0] = token;             // j=0: 0*N + i
        out_row[token * TOPK + 1] = N_TOKENS + token;  // j=1: 1*N + i
    }
}

extern "C" void kernel_launch(void* const* d_in, const int* in_sizes, int n_in,
                              void* d_out, int out_size, void* d_ws, size_t ws_size,
                              hipStream_t stream) {
    const float* hidden = (const float*)d_in[0];   // [32768, 2560] f32
    const float* weight = (const float*)d_in[1];   // [4, 2560] f32
    float* out = (float*)d_out;                    // idx(int32) | weight(f32) | row_idx(int32)

    dim3 grid(N_TOKENS / TOKENS_PER_BLOCK);        // 256 blocks
    dim3 block(BLOCK_THREADS);                     // 8 waves
    moe_gate_wmma_kernel<<<grid, block, 0, stream>>>(hidden, weight, out);
}